// BiAttention_33921651704196
// MI455X (gfx1250) — compile-verified
//
#include <hip/hip_runtime.h>
#include <math.h>

#define B_   32
#define CL_  2048
#define QL_  256
#define D2_  256
#define GW_  1024           // 4*D2
#define NEGV (-10000000.0f)
#define EPSV (1e-13f)

typedef float v2f __attribute__((ext_vector_type(2)));
typedef float v8f __attribute__((ext_vector_type(8)));

// Async global->LDS staging (gfx1250). Guarded: falls back to plain LDS
// staging if the builtins aren't declared by this toolchain.
#if defined(__has_builtin)
#if __has_builtin(__builtin_amdgcn_global_load_async_to_lds_b128) && \
    __has_builtin(__builtin_amdgcn_s_wait_asynccnt)
#define USE_ASYNC_LDS 1
#endif
#endif

#ifdef USE_ASYNC_LDS
// Exact parameter type per hipcc diagnostic: vector_size(16) int, AS1 / AS3.
typedef int v4i_ __attribute__((vector_size(4 * sizeof(int))));
#define GPTR(p) ((__attribute__((address_space(1))) v4i_*)(p))
#define LPTR(p) ((__attribute__((address_space(3))) v4i_*)(p))
#endif

__device__ __forceinline__ v8f wmma_f32(v2f a, v2f b, v8f c) {
  // D = A(16x4) * B(4x16) + C, full f32.
  return __builtin_amdgcn_wmma_f32_16x16x4_f32(false, a, false, b, (short)0, c,
                                               false, false);
}

// ---------------- K0: qdot[b,n] = dot(q[b,n,:], W[256:512]) -----------------
__global__ __launch_bounds__(256) void k0_qdot(const float* __restrict__ q,
                                               const float* __restrict__ W,
                                               float* __restrict__ qdot) {
  int wave = threadIdx.x >> 5;
  int lane = threadIdx.x & 31;
  int row = blockIdx.x * 8 + wave;           // 0 .. B*QL-1
  const float* pr = q + (size_t)row * D2_;
  const float* wq = W + D2_;
  float acc = 0.f;
#pragma unroll
  for (int t = 0; t < 8; ++t) acc += pr[lane + 32 * t] * wq[lane + 32 * t];
#pragma unroll
  for (int off = 16; off >= 1; off >>= 1) acc += __shfl_xor(acc, off);
  if (lane == 0) qdot[row] = acc;
}

// ---------------- K1: fused S -> softmax -> c2q -> G[:, 0:768] --------------
__global__ __launch_bounds__(64) void k1_main(const float* __restrict__ c,
                                              const float* __restrict__ q,
                                              const float* __restrict__ qmask,
                                              const float* __restrict__ W,
                                              const float* __restrict__ qdot,
                                              float* __restrict__ smax,
                                              float* __restrict__ G) {
  __shared__ float w_s[768];
  __shared__ float qd_s[QL_];
  __shared__ float qm_s[QL_];
  __shared__ float cd_s[32];
  __shared__ float cdp[64];
  __shared__ float qt[256 * 20];      // phase1: [16][260]; phase2: [256][20]
  __shared__ float x_s[2 * 16 * 258]; // per-wave S*qm / attn tile (16 x 256)

  const int t = threadIdx.x;
  const int lane = t & 31;
  const int wv = t >> 5;
  const int h = lane >> 4;        // which K-pair half
  const int ln = lane & 15;       // row (A) / col (B,D) within 16
  const int b = blockIdx.y;
  const int m0 = blockIdx.x * 32;

  // stage W, qdot, query_mask
  for (int i = 0; i < 12; ++i) w_s[t + 64 * i] = W[t + 64 * i];
  for (int i = 0; i < 4; ++i) {
    qd_s[t + 64 * i] = qdot[b * QL_ + t + 64 * i];
    qm_s[t + 64 * i] = qmask[b * QL_ + t + 64 * i];
  }
  __syncthreads();

  // cdot for this tile's 32 rows (2 threads per row)
  {
    int row = t >> 1, half = t & 1;
    const float* pr = c + (size_t)(b * CL_ + m0 + row) * D2_ + half * 128;
    const float* wc = &w_s[half * 128];
    float a = 0.f;
    for (int i = 0; i < 128; ++i) a += pr[i] * wc[i];
    cdp[t] = a;
  }
  __syncthreads();
  if (t < 32) cd_s[t] = cdp[2 * t] + cdp[2 * t + 1];
  __syncthreads();

  const int wrow = m0 + wv * 16;  // first c-row owned by this wave
  const float* cbase = c + (size_t)(b * CL_ + wrow + ln) * D2_;
  float* xw = &x_s[wv * 16 * 258];

  // A fragments held in VGPRs across all 16 N-tiles (j-invariant).
  v2f afr[64];                    // 128 VGPRs; k-loops fully unrolled
#pragma unroll
  for (int ks = 0; ks < 64; ++ks) {
    int k = 4 * ks + 2 * h;
    float2 cr = *(const float2*)(cbase + k);
    afr[ks].x = cr.x * w_s[512 + k];       // scale by w_cq once
    afr[ks].y = cr.y * w_s[512 + k + 1];
  }

  // ---- Phase 1: S = (c*w_cq) @ q^T + cdot + qdot, masked, into LDS ----
  for (int j = 0; j < 16; ++j) {
    __syncthreads();
#ifdef USE_ASYNC_LDS
#pragma unroll
    for (int i = 0; i < 16; ++i) {            // stage q rows j*16..+15
      int fid = i * 64 + t;                   // 1024 float4s
      int n = fid >> 6, k4 = fid & 63;
      __builtin_amdgcn_global_load_async_to_lds_b128(
          GPTR(q + (size_t)(b * QL_ + j * 16 + n) * D2_ + k4 * 4),
          LPTR(&qt[n * 260 + k4 * 4]), 0, 0);
    }
    __builtin_amdgcn_s_wait_asynccnt(0);
#else
    for (int i = 0; i < 16; ++i) {
      int fid = i * 64 + t;
      int n = fid >> 6, k4 = fid & 63;
      float4 v = *(const float4*)(q + (size_t)(b * QL_ + j * 16 + n) * D2_ + k4 * 4);
      *(float4*)&qt[n * 260 + k4 * 4] = v;
    }
#endif
    __syncthreads();
    v8f acc = {0.f, 0.f, 0.f, 0.f, 0.f, 0.f, 0.f, 0.f};
#pragma unroll
    for (int ks = 0; ks < 64; ++ks) {
      int k = 4 * ks + 2 * h;
      float2 qr = *(const float2*)&qt[ln * 260 + k];
      v2f bb; bb.x = qr.x; bb.y = qr.y;
      acc = wmma_f32(afr[ks], bb, acc);
    }
    float qmv = qm_s[j * 16 + ln];
    float qdv = qd_s[j * 16 + ln];
#pragma unroll
    for (int r = 0; r < 8; ++r) {
      int m = r + 8 * h;                      // D layout: M = r / r+8
      float s = acc[r] + cd_s[wv * 16 + m] + qdv;
      xw[m * 258 + j * 16 + ln] = s * qmv;    // x = S * qm
    }
  }

  // ---- Softmax per row (2 lanes per row, shfl_xor(16) combine) ----
  {
    float* xr = &xw[ln * 258];
    int c0 = h * 128;
    float mx = -3.0e38f, mS = -3.0e38f;
    for (int i = 0; i < 128; ++i) {
      float v = xr[c0 + i];
      float qmv = qm_s[c0 + i];
      mx = fmaxf(mx, v);
      mS = fmaxf(mS, v + (1.f - qmv) * NEGV); // masked_S for s_max
    }
    mx = fmaxf(mx, __shfl_xor(mx, 16));
    mS = fmaxf(mS, __shfl_xor(mS, 16));
    float s1 = 0.f, s2 = 0.f;
    for (int i = 0; i < 128; ++i) {
      float e = __expf(xr[c0 + i] - mx);
      float eq = e * qm_s[c0 + i];
      s1 += e; s2 += eq;
      xr[c0 + i] = eq;
    }
    s1 += __shfl_xor(s1, 16);
    s2 += __shfl_xor(s2, 16);
    float inv = 1.f / (s2 + EPSV * s1);       // softmax(x)*m/(sum+eps) algebra
    for (int i = 0; i < 128; ++i) xr[c0 + i] *= inv;
    if (h == 0) smax[b * CL_ + wrow + ln] = mS;
  }

  // attn A fragments into VGPRs (reuses afr's registers; j-invariant)
#pragma unroll
  for (int ks = 0; ks < 64; ++ks) {
    int k = 4 * ks + 2 * h;
    float2 ar = *(const float2*)&xw[ln * 258 + k];
    afr[ks].x = ar.x; afr[ks].y = ar.y;
  }

  // ---- Phase 2: c2q = attn @ q, write G[:, 0:768] ----
  for (int j = 0; j < 16; ++j) {
    __syncthreads();
#ifdef USE_ASYNC_LDS
#pragma unroll
    for (int i = 0; i < 16; ++i) {            // stage q[:, j*16..+15]
      int fid = i * 64 + t;                   // 1024 float4s (256 rows x 4)
      int kk = fid >> 2, d4 = fid & 3;
      __builtin_amdgcn_global_load_async_to_lds_b128(
          GPTR(q + (size_t)(b * QL_ + kk) * D2_ + j * 16 + d4 * 4),
          LPTR(&qt[kk * 20 + d4 * 4]), 0, 0);
    }
    __builtin_amdgcn_s_wait_asynccnt(0);
#else
    for (int i = 0; i < 16; ++i) {
      int fid = i * 64 + t;
      int kk = fid >> 2, d4 = fid & 3;
      float4 v = *(const float4*)(q + (size_t)(b * QL_ + kk) * D2_ + j * 16 + d4 * 4);
      *(float4*)&qt[kk * 20 + d4 * 4] = v;
    }
#endif
    __syncthreads();
    v8f acc = {0.f, 0.f, 0.f, 0.f, 0.f, 0.f, 0.f, 0.f};
#pragma unroll
    for (int ks = 0; ks < 64; ++ks) {
      int k = 4 * ks + 2 * h;
      v2f bb;
      bb.x = qt[k * 20 + ln];
      bb.y = qt[(k + 1) * 20 + ln];
      acc = wmma_f32(afr[ks], bb, acc);
    }
    int d = j * 16 + ln;
#pragma unroll
    for (int r = 0; r < 8; ++r) {
      int m = r + 8 * h;
      int grow = wrow + m;
      float cv = c[(size_t)(b * CL_ + grow) * D2_ + d];
      float c2 = acc[r];
      size_t gb = (size_t)(b * CL_ + grow) * GW_;
      G[gb + d] = cv;
      G[gb + 256 + d] = c2;
      G[gb + 512 + d] = cv * c2;
    }
  }
}

// ---------------- K2: attn_q2c = masked_softmax(s_max, cmask), in place ----
__global__ __launch_bounds__(256) void k2_softmax_c(const float* __restrict__ smax,
                                                    const float* __restrict__ cmask,
                                                    float* __restrict__ attn) {
  __shared__ float red[256];
  int b = blockIdx.x, t = threadIdx.x;
  float xv[8], cmv[8], ev[8];
  float mx = -3.0e38f;
#pragma unroll
  for (int i = 0; i < 8; ++i) {
    int idx = t + 256 * i;
    float s = smax[b * CL_ + idx];
    cmv[i] = cmask[b * CL_ + idx];
    xv[i] = s * cmv[i];
    mx = fmaxf(mx, xv[i]);
  }
  red[t] = mx; __syncthreads();
  for (int off = 128; off >= 1; off >>= 1) {
    if (t < off) red[t] = fmaxf(red[t], red[t + off]);
    __syncthreads();
  }
  mx = red[0]; __syncthreads();
  float s1 = 0.f, s2 = 0.f;
#pragma unroll
  for (int i = 0; i < 8; ++i) {
    ev[i] = __expf(xv[i] - mx);
    s1 += ev[i]; s2 += ev[i] * cmv[i];
  }
  red[t] = s1; __syncthreads();
  for (int off = 128; off >= 1; off >>= 1) {
    if (t < off) red[t] += red[t + off];
    __syncthreads();
  }
  s1 = red[0]; __syncthreads();
  red[t] = s2; __syncthreads();
  for (int off = 128; off >= 1; off >>= 1) {
    if (t < off) red[t] += red[t + off];
    __syncthreads();
  }
  s2 = red[0];
  float inv = 1.f / (s2 + EPSV * s1);
#pragma unroll
  for (int i = 0; i < 8; ++i)
    attn[b * CL_ + t + 256 * i] = ev[i] * cmv[i] * inv;
}

// ---------------- K3: partial q2c over 256-row chunks ----------------------
__global__ __launch_bounds__(256) void k3_q2c_partial(const float* __restrict__ c,
                                                      const float* __restrict__ attn,
                                                      float* __restrict__ part) {
  __shared__ float a_s[256];
  int b = blockIdx.y, chunk = blockIdx.x, t = threadIdx.x;
  int c0 = chunk * 256;
  a_s[t] = attn[b * CL_ + c0 + t];
  __syncthreads();
  float acc = 0.f;
  for (int r = 0; r < 256; ++r)
    acc += a_s[r] * c[(size_t)(b * CL_ + c0 + r) * D2_ + t];
  part[(b * 8 + chunk) * D2_ + t] = acc;
}

__global__ __launch_bounds__(256) void k3b_merge(const float* __restrict__ part,
                                                 float* __restrict__ q2c) {
  int b = blockIdx.x, t = threadIdx.x;
  float a = 0.f;
#pragma unroll
  for (int i = 0; i < 8; ++i) a += part[(b * 8 + i) * D2_ + t];
  q2c[b * D2_ + t] = a;
}

// ---------------- K4: G[:, 768:1024] = c * q2c ------------------------------
__global__ __launch_bounds__(256) void k4_gtail(const float* __restrict__ c,
                                                const float* __restrict__ q2c,
                                                float* __restrict__ G) {
  size_t v = (size_t)blockIdx.x * 256 + threadIdx.x;  // float4 idx over B*CL*64
  int d4 = (int)(v & 63);
  size_t m = (v >> 6) % CL_;
  size_t b = (v >> 6) / CL_;
  float4 cv = *(const float4*)(c + (b * CL_ + m) * D2_ + d4 * 4);
  float4 qv = *(const float4*)(q2c + b * D2_ + d4 * 4);
  float4 o;
  o.x = cv.x * qv.x; o.y = cv.y * qv.y; o.z = cv.z * qv.z; o.w = cv.w * qv.w;
  *(float4*)(G + (b * CL_ + m) * GW_ + 768 + d4 * 4) = o;
}

extern "C" void kernel_launch(void* const* d_in, const int* in_sizes, int n_in,
                              void* d_out, int out_size, void* d_ws, size_t ws_size,
                              hipStream_t stream) {
  (void)in_sizes; (void)n_in; (void)out_size; (void)ws_size;
  const float* c  = (const float*)d_in[0];
  const float* cm = (const float*)d_in[1];
  const float* q  = (const float*)d_in[2];
  const float* qm = (const float*)d_in[3];
  const float* W  = (const float*)d_in[4];
  float* G  = (float*)d_out;
  float* ws = (float*)d_ws;

  float* qdot = ws;                      // B*QL      = 8192
  float* smax = qdot + B_ * QL_;         // B*CL      = 65536 (reused as attn)
  float* part = smax + B_ * CL_;         // B*8*D2    = 65536
  float* q2c  = part + B_ * 8 * D2_;     // B*D2      = 8192

  k0_qdot<<<dim3(B_ * QL_ / 8), dim3(256), 0, stream>>>(q, W, qdot);
  k1_main<<<dim3(CL_ / 32, B_), dim3(64), 0, stream>>>(c, q, qm, W, qdot, smax, G);
  k2_softmax_c<<<dim3(B_), dim3(256), 0, stream>>>(smax, cm, smax);  // in-place
  k3_q2c_partial<<<dim3(8, B_), dim3(256), 0, stream>>>(c, smax, part);
  k3b_merge<<<dim3(B_), dim3(256), 0, stream>>>(part, q2c);
  k4_gtail<<<dim3(B_ * CL_ * 64 / 256), dim3(256), 0, stream>>>(c, q2c, G);
}